// RGCN_24343874633797
// MI455X (gfx1250) — compile-verified
//
#include <hip/hip_runtime.h>

typedef __attribute__((ext_vector_type(2))) float v2f;
typedef __attribute__((ext_vector_type(8))) float v8f;

#define N_NODES 50000
#define N_EDGES 500000
#define NREL    8
#define NBASES  4
#define NCOLS   320   // 4*64 basis-projected cols + 64 root-transform cols

// ---------------------------------------------------------------------------
// Pack basis[B,K,64] + root[K,64] into Wp, stored in WMMA-fragment order:
// K-row pairs interleaved so (W[k][j], W[k+1][j]) is a contiguous float2:
//   Wp[((k>>1)*NCOLS + j)*2 + (k&1)] = W[k][j]
// where W[k][j] = basis[j/64, k, j%64] for j<256, root[k, j-256] otherwise.
// ---------------------------------------------------------------------------
__global__ void pack_weights(const float* __restrict__ basis,
                             const float* __restrict__ root,
                             float* __restrict__ Wp, int K) {
    int idx = blockIdx.x * blockDim.x + threadIdx.x;
    if (idx >= K * NCOLS) return;
    int k = idx / NCOLS, j = idx - k * NCOLS;
    float v;
    if (j < 256) {
        int b = j >> 6, o = j & 63;
        v = basis[(b * K + k) * 64 + o];
    } else {
        v = root[k * 64 + (j - 256)];
    }
    Wp[(((k >> 1) * NCOLS) + j) * 2 + (k & 1)] = v;
}

// ---------------------------------------------------------------------------
// C[M,320] = A[M,K] x B[K,320] using V_WMMA_F32_16X16X4_F32 (full f32).
// One wave computes a 16x64 strip: per K-step, 1 b64 A-load + 4 b64 B-loads
// (fragment-packed) feed 4 back-to-back WMMAs.
// ---------------------------------------------------------------------------
__global__ void gemm_wmma_f32(const float* __restrict__ A,
                              const float* __restrict__ Bp,
                              float* __restrict__ C, int M, int K) {
    int gw     = (blockIdx.x * blockDim.x + threadIdx.x) >> 5;
    int lane   = threadIdx.x & 31;
    int mtiles = M >> 4;
    int mtile  = gw / 5;          // 5 n-groups of 64 cover NCOLS=320
    int ngrp   = gw - mtile * 5;
    if (mtile >= mtiles) return;  // uniform per wave -> EXEC stays all-ones

    int row0 = mtile << 4;
    int col0 = ngrp  << 6;
    int m    = lane & 15;
    int kh   = (lane >> 4) << 1;  // lanes 16-31 hold K+2,K+3

    v8f acc0 = {}, acc1 = {}, acc2 = {}, acc3 = {};
    const float* Arow = A + (size_t)(row0 + m) * K;

    for (int k0 = 0; k0 < K; k0 += 4) {
        v2f av = *(const v2f*)(Arow + k0 + kh);   // A[m][k0+kh .. +1]
        // Fragment-packed B: pair (k0+kh, k0+kh+1) is contiguous per column.
        const float* Bf = Bp + ((size_t)((k0 + kh) >> 1) * NCOLS + col0 + m) * 2;
        v2f b0 = *(const v2f*)(Bf + 0);
        v2f b1 = *(const v2f*)(Bf + 32);          // +16 cols
        v2f b2 = *(const v2f*)(Bf + 64);          // +32 cols
        v2f b3 = *(const v2f*)(Bf + 96);          // +48 cols
        acc0 = __builtin_amdgcn_wmma_f32_16x16x4_f32(false, av, false, b0, (short)0, acc0, false, false);
        acc1 = __builtin_amdgcn_wmma_f32_16x16x4_f32(false, av, false, b1, (short)0, acc1, false, false);
        acc2 = __builtin_amdgcn_wmma_f32_16x16x4_f32(false, av, false, b2, (short)0, acc2, false, false);
        acc3 = __builtin_amdgcn_wmma_f32_16x16x4_f32(false, av, false, b3, (short)0, acc3, false, false);
    }

    int rbase = row0 + ((lane >> 4) << 3);  // C/D layout: M = v + (lane/16)*8
    for (int v = 0; v < 8; ++v) {
        float* Cp = C + (size_t)(rbase + v) * NCOLS + col0 + m;
        Cp[0]  = acc0[v];
        Cp[16] = acc1[v];
        Cp[32] = acc2[v];
        Cp[48] = acc3[v];
    }
}

// ---------------------------------------------------------------------------
// Per-(dst,relation) edge counts (shared by both layers).
// ---------------------------------------------------------------------------
__global__ void edge_count(const int* __restrict__ ei,
                           const int* __restrict__ et,
                           float* __restrict__ cnt) {
    int e = blockIdx.x * blockDim.x + threadIdx.x;
    if (e >= N_EDGES) return;
    int dst = ei[N_EDGES + e];
    int t   = et[e];
    atomicAdd(cnt + dst * NREL + t, 1.0f);
}

// ---------------------------------------------------------------------------
// Fused per-edge: gather basis-projected features of src, contract with
// comp[type], divide by the (dst,rel) count (mean fold), scatter-add to agg.
// 64 threads per edge (one per output channel).
// ---------------------------------------------------------------------------
__global__ void edge_msg(const int* __restrict__ ei,
                         const int* __restrict__ et,
                         const float* __restrict__ comp,
                         const float* __restrict__ H,
                         const float* __restrict__ cnt,
                         float* __restrict__ agg) {
    int idx = blockIdx.x * blockDim.x + threadIdx.x;   // < 32M, fits int
    int e = idx >> 6;
    int o = idx & 63;
    if (e >= N_EDGES) return;
    int src = ei[e];
    int dst = ei[N_EDGES + e];
    int t   = et[e];
    const float* hb = H + (size_t)src * NCOLS + o;
    const float* cp = comp + t * NBASES;
    float msg = cp[0] * hb[0]   + cp[1] * hb[64]
              + cp[2] * hb[128] + cp[3] * hb[192];
    float scale = 1.0f / cnt[dst * NREL + t];          // this edge => cnt >= 1
    atomicAdd(agg + dst * 64 + o, msg * scale);
}

// ---------------------------------------------------------------------------
// out = agg + H[:,256:320] (root transform) + bias  [+ ReLU]
// ---------------------------------------------------------------------------
__global__ void finalize(const float* __restrict__ agg,
                         const float* __restrict__ H,
                         const float* __restrict__ bias,
                         float* __restrict__ out, int do_relu) {
    int idx = blockIdx.x * blockDim.x + threadIdx.x;
    if (idx >= N_NODES * 64) return;
    int o = idx & 63;
    int n = idx >> 6;
    float v = agg[idx] + H[(size_t)n * NCOLS + 256 + o] + bias[o];
    if (do_relu) v = fmaxf(v, 0.0f);
    out[idx] = v;
}

// ---------------------------------------------------------------------------
extern "C" void kernel_launch(void* const* d_in, const int* in_sizes, int n_in,
                              void* d_out, int out_size, void* d_ws, size_t ws_size,
                              hipStream_t stream) {
    const float* x      = (const float*)d_in[0];
    const int*   ei     = (const int*)d_in[1];
    const int*   et     = (const int*)d_in[2];
    const float* basis1 = (const float*)d_in[3];
    const float* comp1  = (const float*)d_in[4];
    const float* root1  = (const float*)d_in[5];
    const float* bias1  = (const float*)d_in[6];
    const float* basis2 = (const float*)d_in[7];
    const float* comp2  = (const float*)d_in[8];
    const float* root2  = (const float*)d_in[9];
    const float* bias2  = (const float*)d_in[10];
    float* out = (float*)d_out;

    // Workspace layout (bytes):
    //   Wp : 128*320*4      = 163,840
    //   H  : 50000*320*4    = 64,000,000   (reused by both layers)
    //   cnt: 50000*8*4      = 1,600,000
    //   agg: 50000*64*4     = 12,800,000
    //   h1 : 50000*64*4     = 12,800,000
    char*  ws  = (char*)d_ws;
    float* Wp  = (float*)(ws);
    float* H   = (float*)(ws + 163840);
    float* cnt = (float*)(ws + 163840 + 64000000);
    float* agg = (float*)(ws + 163840 + 64000000 + 1600000);
    float* h1  = (float*)(ws + 163840 + 64000000 + 1600000 + 12800000);

    hipMemsetAsync(cnt, 0, (size_t)N_NODES * NREL * 4, stream);
    hipMemsetAsync(agg, 0, (size_t)N_NODES * 64 * 4, stream);

    const int waves = (N_NODES / 16) * 5;            // 15625 waves
    const int gemm_blocks = (waves * 32 + 255) / 256;

    // ---- Layer 1 ----
    pack_weights<<<(128 * NCOLS + 255) / 256, 256, 0, stream>>>(basis1, root1, Wp, 128);
    gemm_wmma_f32<<<gemm_blocks, 256, 0, stream>>>(x, Wp, H, N_NODES, 128);
    edge_count<<<(N_EDGES + 255) / 256, 256, 0, stream>>>(ei, et, cnt);
    edge_msg<<<(N_EDGES * 64) / 256, 256, 0, stream>>>(ei, et, comp1, H, cnt, agg);
    finalize<<<(N_NODES * 64 + 255) / 256, 256, 0, stream>>>(agg, H, bias1, h1, 1);

    // ---- Layer 2 ----
    hipMemsetAsync(agg, 0, (size_t)N_NODES * 64 * 4, stream);
    pack_weights<<<(64 * NCOLS + 255) / 256, 256, 0, stream>>>(basis2, root2, Wp, 64);
    gemm_wmma_f32<<<gemm_blocks, 256, 0, stream>>>(h1, Wp, H, N_NODES, 64);
    edge_msg<<<(N_EDGES * 64) / 256, 256, 0, stream>>>(ei, et, comp2, H, cnt, agg);
    finalize<<<(N_NODES * 64 + 255) / 256, 256, 0, stream>>>(agg, H, bias2, out, 0);
}